// NaiveS4Layer_10282151707169
// MI455X (gfx1250) — compile-verified
//
#include <hip/hip_runtime.h>
#include <math.h>

typedef __attribute__((ext_vector_type(2))) float v2f;
typedef __attribute__((ext_vector_type(8))) float v8f;

#define Hc 512
#define Nc 64
#define Lc 2048
#define STEPF 1.0f

// workspace layout (float offsets)
#define OFF_ABAR 0         // 64*64
#define OFF_S    4096      // 64*64  (Abar^64)
#define OFF_BL   8192      // 64*64
#define OFF_BBAR 12288     // H*N
#define OFF_P    45056     // 64 * 64*64   (P_i = Abar^i, i=0..63)
#define OFF_Q    307200    // 64 * H * 64  (q[i,h,n] = (C_h^T P_i)[n])
#define OFF_X    2404352   // 32 * 64 * H  (X[j,n,h] = (S^j Bbar_h)[n])
#define OFF_K    3452928   // H * L

// ---------------- f32 WMMA 16x16x4 helpers (wave32) ----------------
// A 16x4 f32: lane m=lane&15, VGPR pair holds K = 2*(lane>>4) + {0,1}
// B 4x16 f32: lane n=lane&15, VGPR pair holds K = 2*(lane>>4) + {0,1}
// C/D 16x16 f32: VGPR r -> M = r + 8*(lane>>4), N = lane&15
__device__ __forceinline__ v2f load_frag_a(const float* A, int rs, int cs,
                                           int m0, int k0, int lane) {
  int m = m0 + (lane & 15);
  int k = k0 + ((lane >> 4) << 1);
  v2f a;
  a.x = A[m * rs + k * cs];
  a.y = A[m * rs + (k + 1) * cs];
  return a;
}
__device__ __forceinline__ v2f load_frag_b(const float* B, int rs, int cs,
                                           int k0, int n0, int lane) {
  int n = n0 + (lane & 15);
  int k = k0 + ((lane >> 4) << 1);
  v2f b;
  b.x = B[k * rs + n * cs];
  b.y = B[(k + 1) * rs + n * cs];
  return b;
}
__device__ __forceinline__ void store_frag_c(float* C, int rs, int cs,
                                             int m0, int n0, int lane, v8f c) {
  int n = n0 + (lane & 15);
  int mb = m0 + ((lane >> 4) << 3);
#pragma unroll
  for (int r = 0; r < 8; ++r) C[(mb + r) * rs + n * cs] = c[r];
}
__device__ __forceinline__ v8f wmma_f32(v2f a, v2f b, v8f c) {
  return __builtin_amdgcn_wmma_f32_16x16x4_f32(false, a, false, b, (short)0, c,
                                               false, false);
}

// ---------------- Kernel 1: build A, invert (ones - A/2), Abar, Bbar ----------------
__global__ void k_setup(const float* __restrict__ Bin, float* __restrict__ ws) {
  __shared__ float aug[64][129];   // [M1 | I] -> [I | BL]
  __shared__ float M2s[64][65];
  __shared__ float colk[64];
  __shared__ int s_piv;
  const int tid = threadIdx.x;  // 256 threads

  // HiPPO A with the eye(1)-broadcast quirk: M1 = 1 - A/2, M2 = 1 + A/2 (elementwise +1)
  for (int e = tid; e < 64 * 64; e += 256) {
    int i = e >> 6, j = e & 63;
    float a = 0.0f;
    if (j < i) {
      float pi = sqrtf(2.0f * (float)i + 3.0f);
      float pj = sqrtf(2.0f * (float)j + 3.0f);
      a = -(pi * pj);
    } else if (j == i) {
      a = -((float)i + 2.0f);
    }
    aug[i][j] = 1.0f - 0.5f * STEPF * a;
    M2s[i][j] = 1.0f + 0.5f * STEPF * a;
    aug[i][64 + j] = (i == j) ? 1.0f : 0.0f;
  }
  __syncthreads();

  // Gauss-Jordan with partial pivoting
  for (int k = 0; k < 64; ++k) {
    if (tid == 0) {
      int best = k;
      float bv = fabsf(aug[k][k]);
      for (int r = k + 1; r < 64; ++r) {
        float v = fabsf(aug[r][k]);
        if (v > bv) { bv = v; best = r; }
      }
      s_piv = best;
    }
    __syncthreads();
    int piv = s_piv;
    if (piv != k) {
      for (int c = tid; c < 128; c += 256) {
        float t = aug[k][c]; aug[k][c] = aug[piv][c]; aug[piv][c] = t;
      }
    }
    __syncthreads();
    float pinv = 1.0f / aug[k][k];
    for (int c = tid; c < 128; c += 256) aug[k][c] *= pinv;
    __syncthreads();
    if (tid < 64) colk[tid] = aug[tid][k];
    __syncthreads();
    for (int e = tid; e < 64 * 128; e += 256) {
      int r = e >> 7, c = e & 127;
      if (r != k) aug[r][c] -= colk[r] * aug[k][c];
    }
    __syncthreads();
  }

  // Abar = BL @ M2 ; also persist BL
  float* Abar = ws + OFF_ABAR;
  float* BLg = ws + OFF_BL;
  for (int e = tid; e < 4096; e += 256) {
    int n = e >> 6, m = e & 63;
    float acc = 0.0f;
    for (int t = 0; t < 64; ++t) acc += aug[n][64 + t] * M2s[t][m];
    Abar[e] = acc;
    BLg[e] = aug[n][64 + m];
  }
  // Bbar[h,n] = STEP * sum_m BL[n,m] * B[h,m]
  float* Bbar = ws + OFF_BBAR;
  for (int e = tid; e < Hc * 64; e += 256) {
    int h = e >> 6, n = e & 63;
    float acc = 0.0f;
    for (int m = 0; m < 64; ++m) acc += aug[n][64 + m] * Bin[h * 64 + m];
    Bbar[e] = acc * STEPF;
  }
}

// ---------------- Kernel 2: P_i = Abar^i (i=0..63), S = Abar^64 ----------------
__global__ void k_powers(float* __restrict__ ws) {
  __shared__ float Ab[64][65];
  __shared__ float b0[64][65];
  __shared__ float b1[64][65];
  const int tid = threadIdx.x;  // 256
  const float* Abar = ws + OFF_ABAR;
  float* P = ws + OFF_P;
  for (int e = tid; e < 4096; e += 256) Ab[e >> 6][e & 63] = Abar[e];
  for (int e = tid; e < 4096; e += 256) {
    int i = e >> 6, j = e & 63;
    float v = (i == j) ? 1.0f : 0.0f;
    b0[i][j] = v;
    P[e] = v;  // P_0 = I
  }
  __syncthreads();
  for (int step = 1; step <= 64; ++step) {
    float(*cur)[65] = (step & 1) ? b0 : b1;
    float(*nxt)[65] = (step & 1) ? b1 : b0;
    for (int e = tid; e < 4096; e += 256) {
      int i = e >> 6, j = e & 63;
      float acc = 0.0f;
      for (int t = 0; t < 64; ++t) acc += cur[i][t] * Ab[t][j];
      nxt[i][j] = acc;
    }
    __syncthreads();
    float* dst = (step < 64) ? (P + step * 4096) : (ws + OFF_S);
    for (int e = tid; e < 4096; e += 256) dst[e] = nxt[e >> 6][e & 63];
    __syncthreads();
  }
}

// ---------------- Kernel 3: Q[i,h,n] = sum_m C[h,m] * P_i[m,n]  (WMMA) ----------------
__global__ void k_q(const float* __restrict__ Cin, float* __restrict__ ws) {
  const int i = blockIdx.x;       // 0..63
  const int h0 = blockIdx.y * 16; // 32 h-tiles
  const int lane = threadIdx.x & 31, wave = threadIdx.x >> 5;  // 4 waves
  const float* Pi = ws + OFF_P + i * 4096;
  const int n0 = wave * 16;
  v8f c = {0.f, 0.f, 0.f, 0.f, 0.f, 0.f, 0.f, 0.f};
#pragma unroll 4
  for (int k = 0; k < 64; k += 4) {
    v2f a = load_frag_a(Cin, 64, 1, h0, k, lane);  // A = C (H x N), M=h, K=m
    v2f b = load_frag_b(Pi, 64, 1, k, n0, lane);   // B = P_i, K=m, N=n
    c = wmma_f32(a, b, c);
  }
  store_frag_c(ws + OFF_Q + i * (Hc * 64), 64, 1, h0, n0, lane, c);
}

// ---------------- Kernel 4: X[j,:,h] = S^j * Bbar[h]  (WMMA, LDS-resident) ----------------
__global__ void k_x(float* __restrict__ ws) {
  __shared__ float Sls[64][66];
  __shared__ float Xa[64][66];
  __shared__ float Xb[64][66];
  const int tid = threadIdx.x;  // 128
  const int lane = tid & 31, wave = tid >> 5;
  const int h0 = blockIdx.x * 64;  // 8 blocks
  const float* Sg = ws + OFF_S;
  const float* Bbar = ws + OFF_BBAR;
  float* Xg = ws + OFF_X;

  for (int e = tid; e < 4096; e += 128) Sls[e >> 6][e & 63] = Sg[e];
  for (int e = tid; e < 4096; e += 128) {
    int n = e >> 6, hl = e & 63;
    Xa[n][hl] = Bbar[(h0 + hl) * 64 + n];  // X_0 = Bbar^T block
  }
  __syncthreads();

  float(*cur)[66] = Xa;
  float(*nxt)[66] = Xb;
  for (int j = 0; j < 32; ++j) {
    float* dst = Xg + j * (64 * Hc);
    for (int e = tid; e < 4096; e += 128) {
      int n = e >> 6, hl = e & 63;
      dst[n * Hc + h0 + hl] = cur[n][hl];
    }
    if (j < 31) {
      // nxt = S * cur : 16 tiles; wave -> row tile, q -> col tile
      const int m0 = wave * 16;
#pragma unroll
      for (int q = 0; q < 4; ++q) {
        const int c0 = q * 16;
        v8f c = {0.f, 0.f, 0.f, 0.f, 0.f, 0.f, 0.f, 0.f};
#pragma unroll 4
        for (int k = 0; k < 64; k += 4) {
          v2f a = load_frag_a(&Sls[0][0], 66, 1, m0, k, lane);
          v2f b = load_frag_b(&cur[0][0], 66, 1, k, c0, lane);
          c = wmma_f32(a, b, c);
        }
        store_frag_c(&nxt[0][0], 66, 1, m0, c0, lane, c);
      }
      __syncthreads();
      float(*tmp)[66] = cur; cur = nxt; nxt = tmp;
    }
  }
}

// ---------------- Kernel 5: K[h,64j+i] = sum_n Q[i,h,n] * X[j,n,h]  (WMMA) ----------------
__global__ void k_kern(float* __restrict__ ws) {
  const int h = blockIdx.x;  // 512 blocks
  const int lane = threadIdx.x & 31, wave = threadIdx.x >> 5;  // 4 waves
  const float* Qb = ws + OFF_Q + h * 64;  // Q[i,h,n]: i stride 32768, n stride 1
  const float* Xb = ws + OFF_X + h;       // X[j,n,h]: n stride 512, j stride 32768
  float* Kg = ws + OFF_K + h * Lc;
  const int i0 = wave * 16;  // 4 i-tiles
#pragma unroll
  for (int t = 0; t < 2; ++t) {
    const int j0 = t * 16;  // 2 j-tiles
    v8f c = {0.f, 0.f, 0.f, 0.f, 0.f, 0.f, 0.f, 0.f};
#pragma unroll 4
    for (int k = 0; k < 64; k += 4) {
      v2f a = load_frag_a(Qb, Hc * 64, 1, i0, k, lane);   // M=i, K=n
      v2f b = load_frag_b(Xb, Hc, 64 * Hc, k, j0, lane);  // K=n, N=j
      c = wmma_f32(a, b, c);
    }
    // K[h, 64*(j0+N) + (i0+M)] : M stride 1, N stride 64
    store_frag_c(Kg, 1, 64, i0, j0, lane, c);
  }
}

// ---------------- Kernel 6: out[h,t] = sum_{s<=t} K[h,s] * y[h,t-s] ----------------
__global__ void k_conv(const float* __restrict__ y, const float* __restrict__ ws,
                       float* __restrict__ out) {
  __shared__ float kk[Lc];
  __shared__ float yy[Lc];
  const int h = blockIdx.x;  // 512 blocks
  const int tid = threadIdx.x;  // 256
  const float* Kg = ws + OFF_K + h * Lc;
  const float* yg = y + h * Lc;
  for (int e = tid; e < Lc; e += 256) { kk[e] = Kg[e]; yy[e] = yg[e]; }
  __syncthreads();
#pragma unroll
  for (int c = 0; c < Lc / 256; ++c) {
    int t = tid + 256 * c;
    float acc = 0.0f;
    for (int s = 0; s <= t; ++s) acc += kk[s] * yy[t - s];
    out[h * Lc + t] = acc;
  }
}

extern "C" void kernel_launch(void* const* d_in, const int* in_sizes, int n_in,
                              void* d_out, int out_size, void* d_ws, size_t ws_size,
                              hipStream_t stream) {
  (void)in_sizes; (void)n_in; (void)out_size; (void)ws_size;
  const float* y = (const float*)d_in[0];   // (H, L)
  const float* B = (const float*)d_in[1];   // (H, N, 1)
  const float* C = (const float*)d_in[2];   // (H, 1, N)
  float* out = (float*)d_out;               // (H, L)
  float* ws = (float*)d_ws;                 // needs ~18 MB

  k_setup<<<1, 256, 0, stream>>>(B, ws);
  k_powers<<<1, 256, 0, stream>>>(ws);
  k_q<<<dim3(64, 32), 128, 0, stream>>>(C, ws);
  k_x<<<8, 128, 0, stream>>>(ws);
  k_kern<<<512, 128, 0, stream>>>(ws);
  k_conv<<<512, 256, 0, stream>>>(y, ws, out);
}